// Layer1_23149873725813
// MI455X (gfx1250) — compile-verified
//
#include <hip/hip_runtime.h>
#include <cstdint>

#define BATCH 64
#define NS    1024
#define NT    20
#define XROW  (2 + 2 * NT)   // 42

typedef __attribute__((ext_vector_type(2))) float v2f;
typedef __attribute__((ext_vector_type(8))) float v8f;

// ---------------------------------------------------------------------------
// Kernel 0: sv, sa (ddof=1 std over all B*T elems) -> a, b, G constants
// ---------------------------------------------------------------------------
__global__ __launch_bounds__(256) void prep_scalars(const float* __restrict__ x,
                                                    const float* __restrict__ w1p,
                                                    const float* __restrict__ w2p,
                                                    float* __restrict__ scal) {
    __shared__ float s0[256], s1[256], s2[256], s3[256];
    int tid = threadIdx.x;
    float vs = 0.f, vq = 0.f, as_ = 0.f, aq = 0.f;
    for (int idx = tid; idx < BATCH * NT; idx += 256) {
        int b = idx / NT, t = idx % NT;
        float vv = x[b * XROW + 2 + t];
        float va = x[b * XROW + 2 + NT + t];
        vs += vv; vq += vv * vv; as_ += va; aq += va * va;
    }
    s0[tid] = vs; s1[tid] = vq; s2[tid] = as_; s3[tid] = aq;
    __syncthreads();
    for (int s = 128; s > 0; s >>= 1) {
        if (tid < s) { s0[tid] += s0[tid + s]; s1[tid] += s1[tid + s];
                       s2[tid] += s2[tid + s]; s3[tid] += s3[tid + s]; }
        __syncthreads();
    }
    if (tid == 0) {
        const float n = (float)(BATCH * NT);
        float var_v = (s1[0] - s0[0] * s0[0] / n) / (n - 1.0f);
        float var_a = (s3[0] - s2[0] * s2[0] / n) / (n - 1.0f);
        float sv = sqrtf(fmaxf(var_v, 0.0f));
        float sa = sqrtf(fmaxf(var_a, 0.0f));
        float w1 = *w1p, w2 = *w2p;
        const float LOG_SQRT_2PI = 0.91893853320467274f;
        float c1 = -logf(sv) - LOG_SQRT_2PI;
        float c2 = -logf(sa) - LOG_SQRT_2PI;
        scal[0] = w1 / (2.0f * sv * sv);   // a
        scal[1] = w2 / (2.0f * sa * sa);   // b
        scal[2] = w1 * c1 + w2 * c2;       // G
    }
}

// ---------------------------------------------------------------------------
// Kernel 1: per-batch start node = argmin d2 with LAST-index tie-break
// ---------------------------------------------------------------------------
__global__ __launch_bounds__(256) void find_start(const float* __restrict__ x,
                                                  const float* __restrict__ node_xy,
                                                  int* __restrict__ start) {
    __shared__ float sv[256]; __shared__ int si[256];
    int b = blockIdx.x, tid = threadIdx.x;
    float sx = x[b * XROW + 0], sy = x[b * XROW + 1];
    float best; int bi;
    {
        int j = tid * 4;
        float dx = node_xy[2 * j] / 2.5f - sx;
        float dy = node_xy[2 * j + 1] / 2.5f - sy;
        best = dx * dx + dy * dy; bi = j;
    }
    for (int q = 1; q < 4; ++q) {
        int j = tid * 4 + q;
        float dx = node_xy[2 * j] / 2.5f - sx;
        float dy = node_xy[2 * j + 1] / 2.5f - sy;
        float v = dx * dx + dy * dy;
        if (v < best || (v == best && j > bi)) { best = v; bi = j; }
    }
    sv[tid] = best; si[tid] = bi;
    __syncthreads();
    for (int s = 128; s > 0; s >>= 1) {
        if (tid < s) {
            float ov = sv[tid + s]; int oi = si[tid + s];
            if (ov < sv[tid] || (ov == sv[tid] && oi > si[tid])) { sv[tid] = ov; si[tid] = oi; }
        }
        __syncthreads();
    }
    if (tid == 0) start[b] = si[0];
}

// ---------------------------------------------------------------------------
// Kernel 2: E[i][j] = {D, TH, C, 0}. Cross term nx_i*nx_j + ny_i*ny_j via
// V_WMMA_F32_16X16X4_F32 (rank-2 Gram matrix, one 16x16 tile per wave).
// ---------------------------------------------------------------------------
__global__ __launch_bounds__(256) void build_E(const float* __restrict__ node_xy,
                                               const int* __restrict__ gid,
                                               float4* __restrict__ E) {
    int wave = (blockIdx.x * 256 + threadIdx.x) >> 5;   // 4096 tiles
    int lane = threadIdx.x & 31;
    int i0 = (wave >> 6) * 16;
    int j0 = (wave & 63) * 16;

    // A (16x4): lanes 0-15 hold K=0,1 = {nx_i, ny_i}; lanes 16-31 hold K=2,3 = 0
    // B (4x16): lanes 0-15 hold K=0,1 rows = {nx_j, ny_j}; lanes 16-31 = 0
    v2f A = {0.0f, 0.0f}, Bm = {0.0f, 0.0f};
    if (lane < 16) {
        A.x  = node_xy[2 * (i0 + lane)];     A.y  = node_xy[2 * (i0 + lane) + 1];
        Bm.x = node_xy[2 * (j0 + lane)];     Bm.y = node_xy[2 * (j0 + lane) + 1];
    }
    v8f c = {};
    c = __builtin_amdgcn_wmma_f32_16x16x4_f32(false, A, false, Bm,
                                              (short)0, c, false, false);

    int jj  = j0 + (lane & 15);
    int ihi = (lane >> 4) << 3;
    float nxj = node_xy[2 * jj], nyj = node_xy[2 * jj + 1];
    float sqj = nxj * nxj + nyj * nyj;
    int gj = gid[jj];
#pragma unroll
    for (int r = 0; r < 8; ++r) {
        int ii = i0 + ihi + r;
        float nxi = node_xy[2 * ii], nyi = node_xy[2 * ii + 1];
        float sqi = nxi * nxi + nyi * nyi;
        float D, TH;
        if (ii == jj) { D = 0.0f; TH = 0.0f; }
        else {
            float d2 = fmaxf(fmaf(-2.0f, c[r], sqi + sqj), 0.0f);
            D  = sqrtf(d2);
            TH = atan2f(nyi - nyj, nxi - nxj);
        }
        float C = (gid[ii] == gj) ? 1.0f : 0.0f;
        E[(size_t)ii * NS + jj] = make_float4(D, TH, C, 0.0f);
    }
}

// ---------------------------------------------------------------------------
// Kernel 3: g0[b][j] = F0[b, j, start_b]  (seed delta)
// ---------------------------------------------------------------------------
__global__ __launch_bounds__(256) void g0_kernel(const float4* __restrict__ E,
                                                 const float* __restrict__ x,
                                                 const int* __restrict__ start,
                                                 const float* __restrict__ scal,
                                                 float* __restrict__ delta0) {
    int b = blockIdx.x;
    int sb = start[b];
    float a = scal[0], bb = scal[1], G = scal[2];
    float vv0 = x[b * XROW + 2], va0 = x[b * XROW + 2 + NT];
    for (int j = threadIdx.x; j < NS; j += blockDim.x) {
        float4 e = E[(size_t)j * NS + sb];
        float q1 = vv0 - e.x, q2 = va0 - e.y;
        float f  = fmaf(-a * q1, q1, fmaf(-bb * q2, q2, G));
        delta0[(size_t)b * NS + j] = e.z * f;
    }
}

// ---------------------------------------------------------------------------
// Kernel 4: max-plus step. One block per 8-row i-tile, all 64 batches.
// Double-buffered async global->LDS staging: while computing chunk k, the
// async engine fills chunk k+1 (ASYNCcnt is in-order per wave, so
// s_wait_asynccnt 3 == "chunk k's loads are done, chunk k+1 may be in flight").
// ---------------------------------------------------------------------------
#define TILE_I 8
#define CJ     64
#define NCHUNK (NS / CJ)

__global__ __launch_bounds__(512) void step_kernel(const float4* __restrict__ E,
                                                   const float* __restrict__ din,
                                                   float* __restrict__ dout,
                                                   const float* __restrict__ x,
                                                   const float* __restrict__ scal,
                                                   int t) {
    // sE row stride 65 f4 -> banks 4*il+4*j (conflict-free across il)
    // sD row stride 68 f  -> banks 4*b+j   (conflict-free across b), 272B rows (16B aligned)
    __shared__ __align__(16) float4 sE[2][TILE_I][CJ + 1];
    __shared__ __align__(16) float  sD[2][BATCH][CJ + 4];

    const int tid = threadIdx.x;
    const int il  = tid & (TILE_I - 1);
    const int b   = tid >> 3;                  // 0..63
    const int i0  = blockIdx.x * TILE_I;
    const float a = scal[0], bb = scal[1], G = scal[2];
    const float vvt = x[b * XROW + 2 + t];
    const float vat = x[b * XROW + 2 + NT + t];

    float m = -__builtin_huge_valf();

    // issue one chunk's async loads into LDS buffer `buf` (3 b128 per thread)
    auto issue = [&](int buf, int j0) {
        {   // E tile: 8 x 64 float4 = 512 (1/thread)
            int r = tid >> 6, cc = tid & 63;
            uint32_t ldst = (uint32_t)(uintptr_t)(&sE[buf][r][cc]);
            unsigned long long src =
                (unsigned long long)(uintptr_t)(E + ((size_t)(i0 + r) * NS + j0 + cc));
            asm volatile("global_load_async_to_lds_b128 %0, %1, off"
                         :: "v"(ldst), "v"(src) : "memory");
        }
#pragma unroll
        for (int k = 0; k < 2; ++k) {   // delta chunk: 64 x 64 floats = 1024 f4 (2/thread)
            int idx = tid + k * 512;
            int r = idx >> 4, c4 = idx & 15;
            uint32_t ldst = (uint32_t)(uintptr_t)(&sD[buf][r][c4 * 4]);
            unsigned long long src =
                (unsigned long long)(uintptr_t)(din + (size_t)r * NS + j0 + c4 * 4);
            asm volatile("global_load_async_to_lds_b128 %0, %1, off"
                         :: "v"(ldst), "v"(src) : "memory");
        }
    };

    issue(0, 0);   // prologue: fill buffer 0
    for (int c = 0; c < NCHUNK; ++c) {
        const int buf = c & 1;
        if (c + 1 < NCHUNK) {
            issue(buf ^ 1, (c + 1) * CJ);                 // prefetch next chunk
            asm volatile("s_wait_asynccnt 0x3" ::: "memory");  // chunk c complete
        } else {
            asm volatile("s_wait_asynccnt 0x0" ::: "memory");
        }
        __syncthreads();   // publish LDS writes across all waves

#pragma unroll 4
        for (int j = 0; j < CJ; ++j) {
            float4 e = sE[buf][il][j];
            float dd = sD[buf][b][j];
            float q1 = vvt - e.x;
            float q2 = vat - e.y;
            float f  = fmaf(-a * q1, q1, fmaf(-bb * q2, q2, G));
            float v  = fmaf(e.z, f, dd);
            m = fmaxf(m, v);
        }
        __syncthreads();   // all consumers done before buffer reuse
    }
    dout[(size_t)b * NS + i0 + il] = m;
}

// ---------------------------------------------------------------------------
// Kernel 5: per-batch max + FIRST-index argmax over new delta
// ---------------------------------------------------------------------------
__global__ __launch_bounds__(256) void reduce_kernel(const float* __restrict__ delta,
                                                     int* __restrict__ gout,
                                                     float* __restrict__ mout,
                                                     int t) {
    __shared__ float sv[256]; __shared__ int si[256];
    int b = blockIdx.x, tid = threadIdx.x;
    float best; int bi;
    {
        int i = tid * 4;
        best = delta[(size_t)b * NS + i]; bi = i;
    }
    for (int q = 1; q < 4; ++q) {
        int i = tid * 4 + q;
        float v = delta[(size_t)b * NS + i];
        if (v > best) { best = v; bi = i; }   // contiguous ascending: > keeps first index
    }
    sv[tid] = best; si[tid] = bi;
    __syncthreads();
    for (int s = 128; s > 0; s >>= 1) {
        if (tid < s) {
            float ov = sv[tid + s]; int oi = si[tid + s];
            if (ov > sv[tid] || (ov == sv[tid] && oi < si[tid])) { sv[tid] = ov; si[tid] = oi; }
        }
        __syncthreads();
    }
    if (tid == 0) { gout[b * NT + t] = si[0]; mout[b * NT + t] = sv[0]; }
}

// ---------------------------------------------------------------------------
// Host launcher
// ---------------------------------------------------------------------------
extern "C" void kernel_launch(void* const* d_in, const int* in_sizes, int n_in,
                              void* d_out, int out_size, void* d_ws, size_t ws_size,
                              hipStream_t stream) {
    const float* x       = (const float*)d_in[0];
    const float* node_xy = (const float*)d_in[1];
    const int*   gid     = (const int*)d_in[2];
    const float* w1      = (const float*)d_in[3];
    const float* w2      = (const float*)d_in[4];

    char* ws = (char*)d_ws;
    // ws layout: [0,64) scalars | [64,320) start | [1024,..) deltaA | deltaB | E (16MB)
    float*  scal  = (float*)(ws + 0);
    int*    start = (int*)(ws + 64);
    float*  dA    = (float*)(ws + 1024);
    float*  dB    = (float*)(ws + 1024 + (size_t)BATCH * NS * 4);
    float4* E     = (float4*)(ws + 1024 + (size_t)2 * BATCH * NS * 4);
    // total need: 1024 + 512KB + 16MB  (~17.3 MB) <= ws_size assumed

    int*   gout = (int*)d_out;                       // gammas (B,T) int32
    float* mout = (float*)d_out + BATCH * NT;        // maxvals (B,T) f32

    prep_scalars<<<1, 256, 0, stream>>>(x, w1, w2, scal);
    find_start<<<BATCH, 256, 0, stream>>>(x, node_xy, start);
    build_E<<<(NS / 16) * (NS / 16) / 8, 256, 0, stream>>>(node_xy, gid, E);
    g0_kernel<<<BATCH, 256, 0, stream>>>(E, x, start, scal, dA);

    float* cur = dA;
    float* nxt = dB;
    for (int t = 0; t < NT; ++t) {
        step_kernel<<<NS / TILE_I, 512, 0, stream>>>(E, cur, nxt, x, scal, t);
        reduce_kernel<<<BATCH, 256, 0, stream>>>(nxt, gout, mout, t);
        float* tmp = cur; cur = nxt; nxt = tmp;
    }
}